// PMLP_14491219656875
// MI455X (gfx1250) — compile-verified
//
#include <hip/hip_runtime.h>

#define N_NODES 100000
#define N_EDGES 3200000
#define NTILES  (N_NODES / 16)   // 6250, exact
#define D_IN    58
#define D_HID   16

typedef __attribute__((ext_vector_type(2))) float v2f;
typedef __attribute__((ext_vector_type(8))) float v8f;

// ---------------------------------------------------------------------------
// Kernel 1: h = x @ W1^T  via V_WMMA_F32_16X16X4_F32 (one 16x16 tile / wave).
// Also zero-initializes hsum and cnt (same index coverage) so every replay of
// the captured graph starts from clean accumulators.
//
// A (16x4 f32, 2 VGPR/lane): lanes 0-15 -> K={k0,k0+1}, lanes 16-31 -> K={k0+2,k0+3},
//                            M = lane&15.
// B (4x16 f32, 2 VGPR/lane): same K striping, N = lane&15; B[k][n] = W1[n][k].
// D (16x16 f32, 8 VGPR):     VGPR v: lanes 0-15 -> M=v, lanes 16-31 -> M=v+8,
//                            N = lane&15.
// Tile guard is wave-uniform => EXEC is all-ones at the WMMA (ISA requirement).
// ---------------------------------------------------------------------------
__global__ __launch_bounds__(256) void pmlp_linear1_wmma(
    const float* __restrict__ x,    // [N, 58]
    const float* __restrict__ W1,   // [16, 58]
    float* __restrict__ h,          // [N, 16]
    float* __restrict__ hsum,       // [N, 16] (zeroed here)
    float* __restrict__ cnt)        // [N]     (zeroed here)
{
  const int lane = threadIdx.x & 31;
  const int wave = threadIdx.x >> 5;
  const int tile = blockIdx.x * 8 + wave;
  if (tile >= NTILES) return;                 // uniform per wave

  const int mn    = lane & 15;                // M for A-frag, N for B/D-frags
  const int khalf = (lane >> 4) * 2;          // 0 or 2

  const float* __restrict__ xrow = x  + (size_t)(tile * 16 + mn) * D_IN;
  const float* __restrict__ wcol = W1 + (size_t)mn * D_IN;   // B[k][n]=W1[n][k]

  v8f acc = {};
  #pragma unroll
  for (int k0 = 0; k0 < 60; k0 += 4) {        // K padded 58 -> 60
    const int ka = k0 + khalf;
    const int kb = ka + 1;
    v2f a, b;
    a.x = (ka < D_IN) ? xrow[ka] : 0.0f;
    a.y = (kb < D_IN) ? xrow[kb] : 0.0f;
    b.x = (ka < D_IN) ? wcol[ka] : 0.0f;
    b.y = (kb < D_IN) ? wcol[kb] : 0.0f;
    // (neg_a, A, neg_b, B, c_mod, C, reuse_a, reuse_b)
    acc = __builtin_amdgcn_wmma_f32_16x16x4_f32(
        false, a, false, b, (short)0, acc, false, false);
  }

  const int moff = (lane >> 4) * 8;
  #pragma unroll
  for (int v = 0; v < 8; ++v) {
    const int r = tile * 16 + v + moff;
    h[(size_t)r * D_HID + mn]    = acc[v];
    hsum[(size_t)r * D_HID + mn] = 0.0f;
  }
  if (lane < 16) cnt[tile * 16 + lane] = 0.0f;
}

// ---------------------------------------------------------------------------
// Kernel 2: edge scatter. 4 threads per edge; each gathers a float4 of h[src]
// (h is L2-resident: 6.4 MB << 192 MB) and does 4 f32 atomic adds to
// hsum[dst]. Lane 0 of each edge bumps the degree counter.
// edge_index is int64 but values < 100000, so read the low dword only.
// ---------------------------------------------------------------------------
__global__ __launch_bounds__(256) void pmlp_scatter(
    const int* __restrict__ ei,     // int64 [2, E] viewed as int32 pairs
    const float* __restrict__ h,    // [N, 16]
    float* __restrict__ hsum,       // [N, 16]
    float* __restrict__ cnt)        // [N]
{
  const long long t = (long long)blockIdx.x * blockDim.x + threadIdx.x;
  const int e = (int)(t >> 2);
  if (e >= N_EDGES) return;
  const int sub = (int)t & 3;
  const int c4  = sub * 4;

  const int src = ei[2 * (size_t)e];                 // edge_index[0][e]
  const int dst = ei[2 * ((size_t)N_EDGES + e)];     // edge_index[1][e]

  const float4 v = *reinterpret_cast<const float4*>(h + (size_t)src * D_HID + c4);
  float* p = hsum + (size_t)dst * D_HID + c4;
  atomicAdd(p + 0, v.x);
  atomicAdd(p + 1, v.y);
  atomicAdd(p + 2, v.z);
  atomicAdd(p + 3, v.w);
  if (sub == 0) atomicAdd(cnt + dst, 1.0f);
}

// ---------------------------------------------------------------------------
// Kernel 3: out[i] = relu(hsum[i]/max(cnt[i],1) + b1) . W2 + b2
// ---------------------------------------------------------------------------
__global__ __launch_bounds__(256) void pmlp_finalize(
    const float* __restrict__ hsum, // [N, 16]
    const float* __restrict__ cnt,  // [N]
    const float* __restrict__ b1,   // [16]
    const float* __restrict__ W2,   // [1, 16]
    const float* __restrict__ b2,   // [1]
    float* __restrict__ out)        // [N]
{
  const int i = blockIdx.x * blockDim.x + threadIdx.x;
  if (i >= N_NODES) return;

  const float inv = 1.0f / fmaxf(cnt[i], 1.0f);
  const float* __restrict__ s = hsum + (size_t)i * D_HID;

  float acc = b2[0];
  #pragma unroll
  for (int j = 0; j < D_HID; ++j) {
    const float hv = fmaxf(fmaf(s[j], inv, b1[j]), 0.0f);
    acc = fmaf(hv, W2[j], acc);
  }
  out[i] = acc;
}

// ---------------------------------------------------------------------------
extern "C" void kernel_launch(void* const* d_in, const int* in_sizes, int n_in,
                              void* d_out, int out_size, void* d_ws, size_t ws_size,
                              hipStream_t stream) {
  const float* x  = (const float*)d_in[0];   // [100000, 58] f32
  const int*   ei = (const int*)d_in[1];     // [2, 3200000] int64 (low dwords)
  const float* W1 = (const float*)d_in[2];   // [16, 58]
  const float* b1 = (const float*)d_in[3];   // [16]
  const float* W2 = (const float*)d_in[4];   // [1, 16]
  const float* b2 = (const float*)d_in[5];   // [1]
  float* out = (float*)d_out;                // [100000]

  float* h    = (float*)d_ws;                        // 6.4 MB
  float* hsum = h    + (size_t)N_NODES * D_HID;      // 6.4 MB
  float* cnt  = hsum + (size_t)N_NODES * D_HID;      // 0.4 MB

  // 1) WMMA GEMM (+ zero accumulators): 8 tiles (waves) per 256-thread block
  pmlp_linear1_wmma<<<(NTILES + 7) / 8, 256, 0, stream>>>(x, W1, h, hsum, cnt);

  // 2) edge scatter: 4 threads per edge
  const long long sthreads = (long long)N_EDGES * 4;
  pmlp_scatter<<<(int)((sthreads + 255) / 256), 256, 0, stream>>>(ei, h, hsum, cnt);

  // 3) mean + bias + relu + final 16->1 linear
  pmlp_finalize<<<(N_NODES + 255) / 256, 256, 0, stream>>>(hsum, cnt, b1, W2, b2, out);
}